// PatchDepatcher_35897336660153
// MI455X (gfx1250) — compile-verified
//
#include <hip/hip_runtime.h>

typedef float v2f __attribute__((ext_vector_type(2)));
typedef float v8f __attribute__((ext_vector_type(8)));

#define B_SZ  16
#define D_K   512
#define P_    16
#define F_    32
#define NP    128                 // FNP / F
#define NE    32                  // 2*P
#define NROWS (B_SZ * F_ * NP)    // 65536
#define WAVES_PER_BLK 8
#define NTILES (NROWS / 16)       // 4096

// ---------------------------------------------------------------------------
// Kernel 1: raw = A @ W^T + b  via V_WMMA_F32_16X16X4_F32, fused epilogue
// (mu passthrough, sigma = clamp(exp(.),1e-6)), scatter to ws[b][n][p][f].
// ---------------------------------------------------------------------------
__global__ __launch_bounds__(256) void gemm_raw_kernel(
    const float* __restrict__ A,    // token_states viewed as [65536][512]
    const float* __restrict__ W,    // [32][512] row-major (e, d)
    const float* __restrict__ bias, // [32]
    float* __restrict__ ws_mu,      // [B][NP][P][F]
    float* __restrict__ ws_sig)     // [B][NP][P][F]
{
    __shared__ float sW[NE * D_K];  // 64 KB of the WGP's 320 KB LDS

    const int tid = threadIdx.x;

    // Cooperative, fully coalesced stage of W into LDS (float4 granularity).
    const float4* Wv  = (const float4*)W;
    float4*       sWv = (float4*)sW;
#pragma unroll
    for (int i = 0; i < (NE * D_K / 4) / 256; ++i)
        sWv[tid + i * 256] = Wv[tid + i * 256];
    __syncthreads();

    const int wave    = tid >> 5;
    const int lane    = tid & 31;
    const int tile    = blockIdx.x * WAVES_PER_BLK + wave;
    const int rowBase = tile * 16;

    // f32 WMMA A-layout: lane = M + 16*(K>>1), vgpr = K&1.
    const int m     = lane & 15;
    const int khalf = lane >> 4;  // 0 or 1
    const float* arow = A + (size_t)(rowBase + m) * D_K + 2 * khalf;
    // B-layout mirrors A with N in place of M: element (k,n) at lane n+16*(k>>1).
    const float* bmu = sW + (size_t)m * D_K + 2 * khalf;          // e = 0..15  (mu)
    const float* bsg = sW + (size_t)(m + 16) * D_K + 2 * khalf;   // e = 16..31 (log sigma)

    v8f cmu = {};
    v8f csg = {};
#pragma unroll 4
    for (int k0 = 0; k0 < D_K; k0 += 4) {
        v2f a  = *(const v2f*)(arow + k0);
        v2f b0 = *(const v2f*)(bmu + k0);
        v2f b1 = *(const v2f*)(bsg + k0);
        // (neg_a, A, neg_b, B, c_mod, C, reuse_a, reuse_b)
        cmu = __builtin_amdgcn_wmma_f32_16x16x4_f32(false, a, false, b0,
                                                    (short)0, cmu, false, false);
        csg = __builtin_amdgcn_wmma_f32_16x16x4_f32(false, a, false, b1,
                                                    (short)0, csg, false, false);
    }

    // C/D layout: lane l, vgpr v holds (M = v + 8*(l>>4), N = l&15).
    const int   e     = lane & 15;          // p index
    const float bmu_s = bias[e];
    const float bsg_s = bias[16 + e];
#pragma unroll
    for (int v = 0; v < 8; ++v) {
        const int row = rowBase + v + 8 * (lane >> 4);
        const int bb  = row >> 12;          // / (F*NP)
        const int f   = (row >> 7) & 31;    // / NP % F
        const int n   = row & 127;          // % NP
        const size_t idx = ((size_t)((bb * NP + n) * P_ + e)) * F_ + f;
        ws_mu[idx] = cmu[v] + bmu_s;
        const float sg = __expf(csg[v] + bsg_s);
        ws_sig[idx] = fmaxf(sg, 1e-6f);
    }
}

// ---------------------------------------------------------------------------
// Kernel 2: overlap-add gather. t = 8n + p has <=2 contributors:
//   (n = t>>3, p = t&7) and (n = (t>>3)-1, p = (t&7)+8). Average by count.
// ---------------------------------------------------------------------------
__global__ __launch_bounds__(256) void combine_kernel(
    const float* __restrict__ ws_mu,
    const float* __restrict__ ws_sig,
    float* __restrict__ out, int T)
{
    const int idx   = blockIdx.x * blockDim.x + threadIdx.x;
    const int total = B_SZ * T * F_;
    if (idx >= total) return;

    const int f  = idx & 31;
    const int rt = idx >> 5;
    const int t  = rt % T;
    const int b  = rt / T;

    const int n1 = t >> 3, p1 = t & 7;
    float smu = 0.f, ssg = 0.f;
    int   c   = 0;
    if (n1 < NP) {
        const size_t i1 = ((size_t)((b * NP + n1) * P_ + p1)) * F_ + f;
        smu += ws_mu[i1]; ssg += ws_sig[i1]; ++c;
    }
    const int n2 = n1 - 1, p2 = p1 + 8;
    if (n2 >= 0 && n2 < NP) {
        const size_t i2 = ((size_t)((b * NP + n2) * P_ + p2)) * F_ + f;
        smu += ws_mu[i2]; ssg += ws_sig[i2]; ++c;
    }
    const float inv = 1.0f / (float)(c > 0 ? c : 1);
    const size_t o = (size_t)b * T * F_ + (size_t)t * F_ + f;
    out[o]                          = smu * inv;   // mu_full
    out[(size_t)B_SZ * T * F_ + o]  = ssg * inv;   // sigma_full
}

extern "C" void kernel_launch(void* const* d_in, const int* in_sizes, int n_in,
                              void* d_out, int out_size, void* d_ws, size_t ws_size,
                              hipStream_t stream)
{
    const float* A    = (const float*)d_in[0];  // token_states
    const float* W    = (const float*)d_in[1];
    const float* bias = (const float*)d_in[2];
    // d_in[3] is T as a device scalar; derive it deterministically from out_size
    // (out_size = 2 * B * T * F) so we stay graph-capture safe.
    const int T = out_size / (2 * B_SZ * F_);

    float* ws_mu  = (float*)d_ws;
    float* ws_sig = ws_mu + (size_t)B_SZ * NP * P_ * F_;  // 4 MB each, 8 MB total

    gemm_raw_kernel<<<NTILES / WAVES_PER_BLK, 256, 0, stream>>>(
        A, W, bias, ws_mu, ws_sig);

    const int total  = B_SZ * T * F_;
    const int blocks = (total + 255) / 256;
    combine_kernel<<<blocks, 256, 0, stream>>>(ws_mu, ws_sig, (float*)d_out, T);
}